// _AuthorStyleEMLTree_45870250721600
// MI455X (gfx1250) — compile-verified
//
#include <hip/hip_runtime.h>
#include <math.h>

typedef float v2f __attribute__((ext_vector_type(2)));
typedef float v8f __attribute__((ext_vector_type(8)));

#define BATCH_N 16384
#define NFEAT   32
#define NLEAF   1024
#define NTERM   33
#define NINT    1023

// ---------------------------------------------------------------------------
// leaf_probs = softmax(leaf_logits / tau) per row (1024 rows x 33 cols).
// One wave32 per row; fp64 math to match fp64 reference, fp32 results.
// ---------------------------------------------------------------------------
__global__ void leaf_softmax_kernel(const double* __restrict__ logits,
                                    const double* __restrict__ tau_p,
                                    float* __restrict__ out_probs,
                                    float* __restrict__ ws_probs) {
    const int row  = blockIdx.x;
    const int lane = threadIdx.x;   // 0..31
    const double tau = tau_p[0];
    const double* rp = logits + (long long)row * NTERM;

    double x0 = rp[lane] / tau;
    double x1 = (lane == 0) ? (rp[32] / tau) : -1.0e300;

    double m = fmax(x0, x1);
    for (int off = 16; off >= 1; off >>= 1)
        m = fmax(m, __shfl_xor(m, off, 32));

    double e0 = exp(x0 - m);
    double e1 = (lane == 0) ? exp(x1 - m) : 0.0;
    double s  = e0 + e1;
    for (int off = 16; off >= 1; off >>= 1)
        s += __shfl_xor(s, off, 32);

    float p0 = (float)(e0 / s);
    out_probs[row * NTERM + lane] = p0;
    ws_probs [row * NTERM + lane] = p0;
    if (lane == 0) {
        float p1 = (float)(e1 / s);
        out_probs[row * NTERM + 32] = p1;
        ws_probs [row * NTERM + 32] = p1;
    }
}

// ---------------------------------------------------------------------------
// gates = sigmoid(blend_logits / tau), fp64 kept in ws for the bypass test.
// ---------------------------------------------------------------------------
__global__ void gate_kernel(const double* __restrict__ blend,
                            const double* __restrict__ tau_p,
                            float* __restrict__ out_g,
                            double* __restrict__ ws_g) {
    const int i = blockIdx.x * blockDim.x + threadIdx.x;
    if (i >= NINT * 2) return;
    double s = 1.0 / (1.0 + exp(-(blend[i] / tau_p[0])));
    out_g[i] = (float)s;
    ws_g[i]  = s;
}

// ---------------------------------------------------------------------------
// current = [1|features] @ leaf_probs^T  via V_WMMA_F32_16X16X4_F32.
// M=16384, N=1024, K=33 (padded to 36 => 9 k-steps of 4).
// One wave per 16x16 tile; 4 waves (128 threads) per block.
// A 16x4 f32 lane layout (ISA 7.12.2): lanes 0-15 hold K=k0,k0+1;
// lanes 16-31 hold K=k0+2,k0+3. B mirrors with N on lanes.
// ---------------------------------------------------------------------------
__global__ void __launch_bounds__(128)
leaf_mix_wmma_kernel(const double* __restrict__ features,     // [BATCH][32] fp64
                     const float*  __restrict__ leaf_probs,   // [NLEAF][NTERM] f32
                     float* __restrict__ current) {           // [BATCH][NLEAF] f32
    const int lane = threadIdx.x & 31;
    const int wave = threadIdx.x >> 5;
    const int tile = blockIdx.x * 4 + wave;      // 65536 tiles total
    const int tm = tile >> 6;                    // 64 tiles along N
    const int tn = tile & 63;
    const int m0 = tm * 16, n0 = tn * 16;
    const int hf  = lane >> 4;                   // which K-half this lane holds
    const int l15 = lane & 15;
    const int mrow = m0 + l15;                   // A row for this lane
    const int ncol = n0 + l15;                   // B col for this lane

    v8f c = {0.f, 0.f, 0.f, 0.f, 0.f, 0.f, 0.f, 0.f};

#pragma unroll
    for (int s = 0; s < 9; ++s) {
        const int kx = s * 4 + hf * 2;
        const int ky = kx + 1;
        // A = cand[mrow][k]:  k==0 -> 1.0, k in [1,32] -> features[k-1], else pad 0
        float ax = (kx == 0) ? 1.0f
                 : ((kx <= 32) ? (float)features[(long long)mrow * NFEAT + (kx - 1)] : 0.0f);
        float ay = (ky <= 32) ? (float)features[(long long)mrow * NFEAT + (ky - 1)] : 0.0f;
        // B[k][n] = leaf_probs[n][k]; pad 0 for k >= 33
        float bx = (kx < NTERM) ? leaf_probs[ncol * NTERM + kx] : 0.0f;
        float by = (ky < NTERM) ? leaf_probs[ncol * NTERM + ky] : 0.0f;
        v2f a = { ax, ay };
        v2f b = { bx, by };
        c = __builtin_amdgcn_wmma_f32_16x16x4_f32(false, a, false, b,
                                                  (short)0, c, false, false);
    }

    // D layout: VGPR r -> (M = m0 + r + 8*hf, N = n0 + l15)
    union { v8f v; float f[8]; } u; u.v = c;
    const int rbase = m0 + hf * 8;
#pragma unroll
    for (int r = 0; r < 8; ++r)
        current[(long long)(rbase + r) * NLEAF + ncol] = u.f[r];
}

// ---------------------------------------------------------------------------
// One tree level: gate blend -> complex exp(L) - log(R) -> sanitize -> clip.
// is_complex==0: input is real-only f32 [B][2*n_pairs] (level-0 current).
// is_complex==1: input is interleaved (re,im) f32 pairs [B][2*n_pairs][2].
// Output is always interleaved complex [B][n_pairs][2].
// ---------------------------------------------------------------------------
__global__ void eml_level_kernel(const float* __restrict__ in, int is_complex,
                                 int n_pairs,
                                 const double* __restrict__ gates, int node_base,
                                 const double* __restrict__ clamp_p,
                                 float* __restrict__ out) {
    const float  clampv = (float)clamp_p[0];
    const double BYP = 1.0 - 2.220446049250313e-16;   // 1 - DBL_EPSILON
    const long long total = (long long)BATCH_N * n_pairs;
    const int ncols = n_pairs * 2;

    for (long long idx = blockIdx.x * (long long)blockDim.x + threadIdx.x;
         idx < total; idx += (long long)gridDim.x * blockDim.x) {
        const int b = (int)(idx / n_pairs);
        const int p = (int)(idx % n_pairs);

        float lre, lim, rre, rim;
        if (is_complex) {
            const float* rowp = in + ((long long)b * ncols) * 2;
            lre = rowp[(2 * p) * 2 + 0]; lim = rowp[(2 * p) * 2 + 1];
            rre = rowp[(2 * p + 1) * 2 + 0]; rim = rowp[(2 * p + 1) * 2 + 1];
        } else {
            const float* rowp = in + (long long)b * ncols;
            lre = rowp[2 * p];     lim = 0.0f;
            rre = rowp[2 * p + 1]; rim = 0.0f;
        }

        const double sl = gates[(long long)(node_base + p) * 2 + 0];
        const double sr = gates[(long long)(node_base + p) * 2 + 1];

        float LR, LI, RR, RI;
        if (sl > BYP) { LR = 1.0f; LI = 0.0f; }
        else { float f = (float)sl; LR = f + (1.0f - f) * lre; LI = (1.0f - f) * lim; }
        if (sr > BYP) { RR = 1.0f; RI = 0.0f; }
        else { float f = (float)sr; RR = f + (1.0f - f) * rre; RI = (1.0f - f) * rim; }

        // exp(L): e^LR * (cos LI + i sin LI)
        float el = expf(LR);
        float sn, cs;
        sincosf(LI, &sn, &cs);
        float eRe = el * cs, eIm = el * sn;
        // log(R): 0.5*ln(|R|^2) + i*atan2(RI, RR)
        float mag2 = RR * RR + RI * RI;
        float lRe = 0.5f * logf(mag2);
        float lIm = atan2f(RI, RR);

        float cre = eRe - lRe;
        float cim = eIm - lIm;
        // nan_to_num (nan->0, +-inf handled by min/max clamp) + clip
        cre = isnan(cre) ? 0.0f : fminf(fmaxf(cre, -clampv), clampv);
        cim = isnan(cim) ? 0.0f : fminf(fmaxf(cim, -clampv), clampv);

        float* op = out + ((long long)b * n_pairs + p) * 2;
        op[0] = cre;
        op[1] = cim;
    }
}

// Copy final level (B x 1 complex) into output slot 0.
__global__ void final_copy_kernel(const float* __restrict__ lvl_last,
                                  float* __restrict__ out0) {
    const int i = blockIdx.x * blockDim.x + threadIdx.x;
    if (i < BATCH_N * 2) out0[i] = lvl_last[i];
}

// ---------------------------------------------------------------------------
extern "C" void kernel_launch(void* const* d_in, const int* in_sizes, int n_in,
                              void* d_out, int out_size, void* d_ws, size_t ws_size,
                              hipStream_t stream) {
    (void)in_sizes; (void)n_in; (void)out_size; (void)ws_size;

    const double* features    = (const double*)d_in[0];
    const double* leaf_logits = (const double*)d_in[1];
    const double* blend       = (const double*)d_in[2];
    const double* tau_leaf    = (const double*)d_in[3];
    const double* tau_gate    = (const double*)d_in[4];
    const double* clampp      = (const double*)d_in[5];
    float* out = (float*)d_out;

    // workspace layout
    float*  ws_current = (float*)d_ws;                                   // 16384*1024 f32 = 64 MB
    float*  ws_leaf    = ws_current + (size_t)BATCH_N * NLEAF;           // 1024*33 f32
    double* ws_gates   = (double*)(ws_leaf + (size_t)NLEAF * NTERM);     // 1023*2 f64 (8B aligned)

    // d_out layout (floats; complex = interleaved re,im), return-tuple order
    const size_t off_out0 = 0;                                // (B,) complex   -> 32768
    const size_t off_leaf = (size_t)BATCH_N * 2;              // leaf_probs     -> 33792
    const size_t off_gate = off_leaf + (size_t)NLEAF * NTERM; // gate_probs     -> 2046
    size_t cur_off = off_gate + (size_t)NINT * 2;             // eml levels start

    leaf_softmax_kernel<<<NLEAF, 32, 0, stream>>>(leaf_logits, tau_leaf,
                                                  out + off_leaf, ws_leaf);
    gate_kernel<<<(NINT * 2 + 255) / 256, 256, 0, stream>>>(blend, tau_gate,
                                                            out + off_gate, ws_gates);

    // WMMA GEMM: 1024 x 64 = 65536 tiles, 4 waves/block -> 16384 blocks
    leaf_mix_wmma_kernel<<<(BATCH_N / 16) * (NLEAF / 16) / 4, 128, 0, stream>>>(
        features, ws_leaf, ws_current);

    // Tree reduction: 10 dependent levels
    const float* in = ws_current;
    int is_complex = 0;
    int node_base  = 0;
    float* last_out = nullptr;
    for (int n_pairs = NLEAF / 2; n_pairs >= 1; n_pairs >>= 1) {
        float* o = out + cur_off;
        long long total = (long long)BATCH_N * n_pairs;
        int blocks = (int)((total + 255) / 256);
        eml_level_kernel<<<blocks, 256, 0, stream>>>(in, is_complex, n_pairs,
                                                     ws_gates, node_base, clampp, o);
        node_base += n_pairs;
        cur_off   += (size_t)BATCH_N * n_pairs * 2;
        in = o;
        is_complex = 1;
        last_out = o;
    }

    final_copy_kernel<<<(BATCH_N * 2 + 255) / 256, 256, 0, stream>>>(last_out,
                                                                     out + off_out0);
}